// TR_Embedding_21182778704895
// MI455X (gfx1250) — compile-verified
//
#include <hip/hip_runtime.h>
#include <hip/hip_bf16.h>

// TR-embedding: out[t, j] = sum_{a,b} core0[a, x[t], b] * core1[b, j, a]
//   x:     (4096*50) int32
//   core0: (16, 200000, 16) f32
//   core1: (16, 256, 16) f32
//   out:   (204800, 256) f32
//
// Strategy: per 16-token group, a (16 x 256) x (256 x 256) GEMM via
// V_WMMA_F32_16X16X4_F32 (fp32 accuracy). core1 is repacked once into d_ws so
// B fragments are single b64 loads; A tiles staged in LDS with a conflict-free
// paired layout so A fragments are single ds_load_b64.

#define TRV 200000
#define TRE 256
#define TRK 256  // 16*16 contraction

typedef __attribute__((ext_vector_type(2))) float v2f;
typedef __attribute__((ext_vector_type(8))) float v8f;

// ---------------------------------------------------------------------------
// Pre-pass: Wt2[(k>>1)*512 + j*2 + (k&1)] = core1[b, j, a],  k = a*16 + b
// ---------------------------------------------------------------------------
__global__ __launch_bounds__(256) void tr_emb_pack_w(
    const float* __restrict__ core1, float* __restrict__ Wt2) {
  int id = blockIdx.x * 256 + threadIdx.x;   // 0..65535
  int k = id >> 8;                           // 0..255
  int j = id & 255;                          // 0..255
  int a = k >> 4;
  int b = k & 15;
  float v = core1[(size_t)b * (TRE * 16) + (size_t)j * 16 + (size_t)a];
  Wt2[(size_t)(k >> 1) * 512 + (size_t)j * 2 + (k & 1)] = v;
}

// ---------------------------------------------------------------------------
// Main kernel: one block = 16 tokens, 8 waves, each wave owns 2 n-strips.
// ---------------------------------------------------------------------------
__global__ __launch_bounds__(256) void tr_emb_main(
    const int* __restrict__ x, const float* __restrict__ core0,
    const float* __restrict__ Wt2, float* __restrict__ out, int ntok) {
  // A tile staged as pairs: Alds[p][tok*2 + c], p = k/2, c = k&1.
  // Row stride 32 dwords -> b64 reads hit disjoint 32-bank halves (no conflict).
  __shared__ float Alds[128 * 32];  // 16 KB

  const int tbase = blockIdx.x * 16;
  const int t = threadIdx.x;

  // ---- stage M[tok][k] = core0[a, idx[tok], b] into LDS ----
  {
    int a = t >> 4;     // 0..15
    int tok = t & 15;   // 0..15
    int gt = tbase + tok;
    int idx = (gt < ntok) ? x[gt] : 0;
    const float* src =
        core0 + (size_t)a * ((size_t)TRV * 16) + (size_t)idx * 16;
#pragma unroll
    for (int m = 0; m < 8; ++m) {             // b = 2m, 2m+1
      float2 v = ((const float2*)src)[m];
      int p = a * 8 + m;                      // (a*16 + 2m)/2
      *(float2*)(&Alds[p * 32 + tok * 2]) = v;
    }
  }
  __syncthreads();

  const int lane = t & 31;
  const int wave = t >> 5;
  const int tok = lane & 15;   // N column within strip / A row (token)
  const int hi = lane >> 4;    // lane half selects K pair 0/1 vs 2/3

  const int nb0 = wave * 32;       // first 16-col strip
  const int nb1 = nb0 + 16;        // second 16-col strip

  // Per-lane bases; per-iteration deltas are compile-time immediates.
  const float* aP = &Alds[hi * 32 + tok * 2];            // + it*64 floats
  const float* bP0 = Wt2 + (size_t)hi * 512 + (size_t)(nb0 + tok) * 2;  // + it*1024
  const float* bP1 = Wt2 + (size_t)hi * 512 + (size_t)(nb1 + tok) * 2;

  v8f acc0 = {};
  v8f acc1 = {};

#pragma unroll
  for (int it = 0; it < 64; ++it) {
    v2f af = *(const v2f*)(aP + it * 64);      // A: M=tok, K = 4it+2hi, +1
    v2f bf0 = *(const v2f*)(bP0 + it * 1024);  // B: K = 4it+2hi,+1  N = nb0+tok
    v2f bf1 = *(const v2f*)(bP1 + it * 1024);
    acc0 = __builtin_amdgcn_wmma_f32_16x16x4_f32(
        false, af, false, bf0, (short)0, acc0, false, false);
    acc1 = __builtin_amdgcn_wmma_f32_16x16x4_f32(
        false, af, false, bf1, (short)0, acc1, false, false);
  }

  // ---- store D: VGPR r -> row (r + 8*hi), lane -> column tok ----
  float* orow = out + (size_t)(tbase + hi * 8) * TRE + tok;
#pragma unroll
  for (int r = 0; r < 8; ++r) {
    int trow = tbase + hi * 8 + r;
    if (trow < ntok) {
      orow[(size_t)r * TRE + nb0] = acc0[r];
      orow[(size_t)r * TRE + nb1] = acc1[r];
    }
  }
}

// ---------------------------------------------------------------------------
extern "C" void kernel_launch(void* const* d_in, const int* in_sizes, int n_in,
                              void* d_out, int out_size, void* d_ws,
                              size_t ws_size, hipStream_t stream) {
  const int* x = (const int*)d_in[0];          // 204800 int32
  const float* core0 = (const float*)d_in[1];  // 16*200000*16 f32
  const float* core1 = (const float*)d_in[2];  // 16*256*16 f32
  float* out = (float*)d_out;                  // 204800*256 f32
  float* Wt2 = (float*)d_ws;                   // 256 KB repacked core1

  (void)n_in;
  (void)out_size;
  (void)ws_size;

  int ntok = in_sizes[0];  // 204800

  // Repack core1 -> Wt2 (pair-interleaved, row-major over j).
  tr_emb_pack_w<<<dim3(256), dim3(256), 0, stream>>>(core1, Wt2);

  // Main WMMA kernel: 16 tokens per block.
  int nblocks = (ntok + 15) / 16;
  tr_emb_main<<<dim3(nblocks), dim3(256), 0, stream>>>(x, core0, Wt2, out,
                                                       ntok);
}